// CMDNet_58274116272706
// MI455X (gfx1250) — compile-verified
//
#include <hip/hip_runtime.h>
#include <hip/hip_bf16.h>

typedef __attribute__((ext_vector_type(2))) float v2f;
typedef __attribute__((ext_vector_type(8))) float v8f;
typedef __attribute__((ext_vector_type(4))) unsigned int u32x4;
typedef __attribute__((ext_vector_type(8))) unsigned int u32x8;

#define BSZ   16384
#define NR    64
#define NT    32
#define MC    4
#define NITER 64

#define WAVES      4    // waves (== batch elements) per workgroup
#define HT_STRIDE  40   // padded LDS row stride (floats): 32 data + 8 pad (TDM pad_amount)
#define HH_STRIDE  40

__device__ __forceinline__ float bcast_lane(float v, int lane) {
    return __builtin_bit_cast(float,
        __builtin_amdgcn_readlane(__builtin_bit_cast(int, v), lane));
}

__global__ __launch_bounds__(WAVES * 32) void cmdnet_fused(
    const float* __restrict__ yt,      // [B,64]
    const float* __restrict__ Ht,      // [B,64,32]
    const float* __restrict__ sigmat0, // [B]
    const float* __restrict__ mconst,  // [4]
    const float* __restrict__ alpha,   // [4]
    const float* __restrict__ taui,    // [65]
    const float* __restrict__ delta,   // [64]
    float* __restrict__ out_ft,        // [B,32,4]
    float* __restrict__ out_xt)        // [B,32]
{
    __shared__ __align__(16) float HtS[WAVES][NR][HT_STRIDE];  // TDM-staged Ht tile
    __shared__ __align__(16) float HHS[WAVES][NT][HH_STRIDE];  // HH = Ht^T Ht
    __shared__ float tauiS[NITER + 1];
    __shared__ float deltaS[NITER];

    const int wave = threadIdx.x >> 5;
    const int lane = threadIdx.x & 31;
    const int b    = blockIdx.x * WAVES + wave;

    // ---- per-block scalar cache: kill per-iteration VMEM inside the 64-step loop ----
    if (threadIdx.x < NITER + 1) tauiS[threadIdx.x] = taui[threadIdx.x];
    {
        int t2 = (int)threadIdx.x - 64;
        if (t2 >= 0 && t2 < NITER) deltaS[t2] = delta[t2];
    }

    // ---------------- TDM: DMA Ht[b] (64x32 f32, 8 KiB) HBM -> LDS ----------------
    // D# (CDNA5 ISA): 2D tensor, data_size=4B, tensor/tile 32x64,
    // dim0 stride 32, LDS pad_enable: +8 DWORDs every 32 DWORDs => row stride 40.
    {
        unsigned long long ga =
            (unsigned long long)(uintptr_t)(Ht + (size_t)b * NR * NT);
        unsigned int ldsa =
            (unsigned int)(uintptr_t)&HtS[wave][0][0];   // low 32 bits = LDS byte addr

        u32x4 g0;
        g0.x = 1u;                                       // count=1, user descriptor
        g0.y = ldsa;                                     // lds_addr
        g0.z = (unsigned int)ga;                         // global_addr[31:0]
        g0.w = ((unsigned int)(ga >> 32) & 0x01FFFFFFu)  // global_addr[56:32]
             | (2u << 30);                               // type = 2 ("image")

        u32x8 g1;
        g1.s0 = (2u << 16)          // data_size = 4 bytes
              | (1u << 20)          // pad_enable
              | (4u << 22)          // pad_interval: every 32 DWORDs
              | (7u << 25);         // pad_amount:   8 DWORDs
        g1.s1 = ((unsigned)NT & 0xFFFFu) << 16;  // tensor_dim0[15:0] = 32
        g1.s2 = ((unsigned)NR & 0xFFFFu) << 16;  // tensor_dim0 hi=0 | tensor_dim1[15:0] = 64
        g1.s3 = ((unsigned)NT) << 16;            // tensor_dim1 hi=0 | tile_dim0 = 32
        g1.s4 = (unsigned)NR;                    // tile_dim1 = 64 | tile_dim2 = 0
        g1.s5 = (unsigned)NT;                    // tensor_dim0_stride[31:0] = 32
        g1.s6 = 0u;                              // stride0 hi | stride1 lo
        g1.s7 = 0u;                              // stride1 hi

        asm volatile("tensor_load_to_lds %0, %1"
                     :: "s"(g0), "s"(g1) : "memory");
        __builtin_amdgcn_s_wait_tensorcnt(0);
    }
    __syncthreads();

    // ---------------- yH[lane] = sum_r yt[b][r] * Ht[b][r][lane] ----------------
    float yv0 = yt[(size_t)b * NR + lane];
    float yv1 = yt[(size_t)b * NR + 32 + lane];
    float yH = 0.0f;
    #pragma unroll
    for (int r = 0; r < 32; ++r)
        yH = fmaf(bcast_lane(yv0, r), HtS[wave][r][lane], yH);
    #pragma unroll
    for (int r = 0; r < 32; ++r)
        yH = fmaf(bcast_lane(yv1, r), HtS[wave][32 + r][lane], yH);

    // ---------------- HH = Ht^T Ht via V_WMMA_F32_16X16X4_F32 ----------------
    // A (16x4 f32) layout: lanes 0-15 -> VGPR0=K0, VGPR1=K1; lanes 16-31 -> K2,K3.
    // For the symmetric product A^T A the A-frag and B-frag register contents
    // coincide, so two fragment loads feed all four 16x16 tiles.
    const int mrow = lane & 15;
    const int khi  = lane >> 4;

    v8f d00 = {}, d01 = {}, d10 = {}, d11 = {};
    #pragma unroll
    for (int k0 = 0; k0 < NR; k0 += 4) {
        int r0 = k0 + 2 * khi;
        v2f f0, f1;
        f0.x = HtS[wave][r0][mrow];          // tile-rows 0..15
        f0.y = HtS[wave][r0 + 1][mrow];
        f1.x = HtS[wave][r0][16 + mrow];     // tile-rows 16..31
        f1.y = HtS[wave][r0 + 1][16 + mrow];
        d00 = __builtin_amdgcn_wmma_f32_16x16x4_f32(false, f0, false, f0, (short)0, d00, false, false);
        d01 = __builtin_amdgcn_wmma_f32_16x16x4_f32(false, f0, false, f1, (short)0, d01, false, false);
        d10 = __builtin_amdgcn_wmma_f32_16x16x4_f32(false, f1, false, f0, (short)0, d10, false, false);
        d11 = __builtin_amdgcn_wmma_f32_16x16x4_f32(false, f1, false, f1, (short)0, d11, false, false);
    }

    // C/D layout: VGPR p -> row p (lanes 0-15) / row p+8 (lanes 16-31), col = lane%16
    #pragma unroll
    for (int p = 0; p < 8; ++p) {
        int rr = p + 8 * khi;
        HHS[wave][rr][mrow]           = d00[p];
        HHS[wave][rr][16 + mrow]      = d01[p];
        HHS[wave][16 + rr][mrow]      = d10[p];
        HHS[wave][16 + rr][16 + mrow] = d11[p];
    }
    __syncthreads();

    // HH is symmetric: column `lane` == row `lane` -> vector b128 reads
    float HHc[NT];
    #pragma unroll
    for (int j = 0; j < 8; ++j) {
        float4 v = *(const float4*)&HHS[wave][lane][j * 4];
        HHc[4 * j + 0] = v.x; HHc[4 * j + 1] = v.y;
        HHc[4 * j + 2] = v.z; HHc[4 * j + 3] = v.w;
    }

    // ---------------- constants ----------------
    const float mv0 = mconst[0], mv1 = mconst[1], mv2 = mconst[2], mv3 = mconst[3];
    const float la0 = __logf(alpha[0]), la1 = __logf(alpha[1]);
    const float la2 = __logf(alpha[2]), la3 = __logf(alpha[3]);
    const float sg   = sigmat0[b];
    const float sig2 = sg * sg;

    float G0 = 0.0f, G1 = 0.0f, G2 = 0.0f, G3 = 0.0f;

    // ---------------- 64 unfolded gradient-descent iterations ----------------
    for (int it = 0; it < NITER; ++it) {
        float ta    = fabsf(tauiS[it]);
        float scale = (it == 0) ? 1.0f : ta;
        float dlt   = deltaS[it];

        // softmax over M=4 (max-subtracted, matches jax.nn.softmax)
        float z0 = (la0 + G0) * scale, z1 = (la1 + G1) * scale;
        float z2 = (la2 + G2) * scale, z3 = (la3 + G3) * scale;
        float mx = fmaxf(fmaxf(z0, z1), fmaxf(z2, z3));
        float e0 = __expf(z0 - mx), e1 = __expf(z1 - mx);
        float e2 = __expf(z2 - mx), e3 = __expf(z3 - mx);
        float inv = 1.0f / (e0 + e1 + e2 + e3);
        float f0 = e0 * inv, f1 = e1 * inv, f2 = e2 * inv, f3 = e3 * inv;

        float xt = f0 * mv0 + f1 * mv1 + f2 * mv2 + f3 * mv3;

        // xHH[lane] = sum_i xt_i * HH[i][lane] : broadcast + FMA (min MAC count)
        float acc = 0.0f;
        #pragma unroll
        for (int i = 0; i < 32; ++i)
            acc = fmaf(bcast_lane(xt, i), HHc[i], acc);
        float tres = ta * (acc - yH);   // fold ta into the residual once

        // grad_L = sig2*(1-exp(-G)) + ta*ft*(m - xt)*(xHH - yH)
        float g0 = fmaf(f0 * (mv0 - xt), tres, sig2 * (1.0f - __expf(-G0)));
        float g1 = fmaf(f1 * (mv1 - xt), tres, sig2 * (1.0f - __expf(-G1)));
        float g2 = fmaf(f2 * (mv2 - xt), tres, sig2 * (1.0f - __expf(-G2)));
        float g3 = fmaf(f3 * (mv3 - xt), tres, sig2 * (1.0f - __expf(-G3)));
        G0 -= dlt * g0; G1 -= dlt * g1; G2 -= dlt * g2; G3 -= dlt * g3;
    }

    // ---------------- last layer: softmax + soft symbol only ----------------
    {
        float taF = fabsf(tauiS[NITER]);
        float z0 = (la0 + G0) * taF, z1 = (la1 + G1) * taF;
        float z2 = (la2 + G2) * taF, z3 = (la3 + G3) * taF;
        float mx = fmaxf(fmaxf(z0, z1), fmaxf(z2, z3));
        float e0 = __expf(z0 - mx), e1 = __expf(z1 - mx);
        float e2 = __expf(z2 - mx), e3 = __expf(z3 - mx);
        float inv = 1.0f / (e0 + e1 + e2 + e3);
        float f0 = e0 * inv, f1 = e1 * inv, f2 = e2 * inv, f3 = e3 * inv;
        float xt = f0 * mv0 + f1 * mv1 + f2 * mv2 + f3 * mv3;

        float4 fo; fo.x = f0; fo.y = f1; fo.z = f2; fo.w = f3;
        ((float4*)out_ft)[(size_t)b * NT + lane] = fo;   // [B,32,4], 16B aligned
        out_xt[(size_t)b * NT + lane] = xt;
    }
}

extern "C" void kernel_launch(void* const* d_in, const int* in_sizes, int n_in,
                              void* d_out, int out_size, void* d_ws, size_t ws_size,
                              hipStream_t stream) {
    (void)in_sizes; (void)n_in; (void)out_size; (void)d_ws; (void)ws_size;
    const float* yt    = (const float*)d_in[0];
    const float* Ht    = (const float*)d_in[1];
    const float* sig   = (const float*)d_in[2];
    const float* m     = (const float*)d_in[3];
    const float* alpha = (const float*)d_in[4];
    const float* taui  = (const float*)d_in[5];
    const float* delta = (const float*)d_in[6];

    float* out_ft = (float*)d_out;                       // [B,32,4] first
    float* out_xt = out_ft + (size_t)BSZ * NT * MC;      // then [B,32]

    dim3 grid(BSZ / WAVES);
    dim3 block(WAVES * 32);
    cmdnet_fused<<<grid, block, 0, stream>>>(yt, Ht, sig, m, alpha, taui, delta,
                                             out_ft, out_xt);
}